// TempoAttentionCore_65816078844473
// MI455X (gfx1250) — compile-verified
//
#include <hip/hip_runtime.h>
#include <hip/hip_bf16.h>

// CDNA5 (gfx1250) fused attention: scores = QK^T/sqrt(D), attn = softmax(scores),
// context = attn @ V.  fp32 WMMA (V_WMMA_F32_16X16X4_F32): kernel is HBM-write
// bound (~536MB score/attn outputs), so full-precision matrix ops are free.
// Outputs use non-temporal b128 stores so the streaming writes don't evict the
// L2-resident K/V working set (32MB << 192MB L2).

typedef __attribute__((ext_vector_type(2))) float v2f;
typedef __attribute__((ext_vector_type(4))) float v4f;
typedef __attribute__((ext_vector_type(8))) float v8f;

#define TBH    64          // T*B*H batch-heads
#define NTOK   1024        // sequence length N
#define DHEAD  64          // head dim D
#define ROWS   16          // query rows per block
#define QSTR   68          // padded LDS row stride for Q tile (bank-conflict free, 16B-aligned rows)
#define SSTR   1028        // padded LDS row stride for score strip (16B-aligned rows)
#define SCALE  0.125f      // 1/sqrt(64)

// LDS layout (floats)
#define LDS_Q  0
#define LDS_S  (ROWS * QSTR)                       // 1088
#define LDS_P  (LDS_S + ROWS * SSTR)               // 1088 + 16448 = 17536
#define LDS_FLOATS (LDS_P + 8 * 256)               // + 2048 = 19584  (~76.5 KB)

__device__ __forceinline__ float wave_max(float v) {
    #pragma unroll
    for (int o = 16; o > 0; o >>= 1) v = fmaxf(v, __shfl_xor(v, o, 32));
    return v;
}
__device__ __forceinline__ float wave_sum(float v) {
    #pragma unroll
    for (int o = 16; o > 0; o >>= 1) v += __shfl_xor(v, o, 32);
    return v;
}

__global__ __launch_bounds__(256)
void tempo_attn_kernel(const float* __restrict__ q,
                       const float* __restrict__ k,
                       const float* __restrict__ v,
                       float* __restrict__ ctx,
                       float* __restrict__ scores,
                       float* __restrict__ attn)
{
    extern __shared__ float smem[];
    const int rt   = blockIdx.x;        // query row-tile 0..63
    const int bh   = blockIdx.y;        // batch-head 0..63
    const int tid  = threadIdx.x;       // 0..255 (8 wave32s)
    const int w    = tid >> 5;
    const int lane = tid & 31;
    const int lh   = lane >> 4;         // half-wave: selects K pair in 16x4 frags
    const int l15  = lane & 15;         // M (A) or N (B) index within tile

    const long long bhND = (long long)bh * NTOK * DHEAD;
    const int row0 = rt * ROWS;

    // ---- stage Q tile (16 x 64) into padded LDS ----
    {
        const v4f* qg = (const v4f*)(q + bhND + (long long)row0 * DHEAD);
        v4f t = qg[tid];                // 256 threads x 16B = 4KB tile
        const int g = tid * 4;
        const int m = g >> 6, c = g & 63;
        *(v4f*)(smem + LDS_Q + m * QSTR + c) = t;   // QSTR*4 and c are 16B-aligned
    }
    __syncthreads();

    // ---- A fragments of Q: 16 K-steps of 16x4 (layout: lane=M, half-wave=K pair) ----
    v2f aq[16];
    {
        const float* qs = smem + LDS_Q + l15 * QSTR + lh * 2;
        #pragma unroll
        for (int kk = 0; kk < 16; ++kk) {
            aq[kk].x = qs[kk * 4 + 0];
            aq[kk].y = qs[kk * 4 + 1];
        }
    }

    // ---- QK^T: wave w owns key-column tiles [8w, 8w+8) ----
    const float* kbase = k + bhND;
    #pragma unroll 1
    for (int kt = 0; kt < 8; ++kt) {
        const int col0 = (w * 8 + kt) * 16;
        const float* kg = kbase + (long long)(col0 + l15) * DHEAD + lh * 2;
        v8f acc = {};
        #pragma unroll
        for (int kk = 0; kk < 16; ++kk) {
            v2f b;                       // B[k][n] = K[n][k] (K row-major => same gather as A)
            b.x = kg[kk * 4 + 0];
            b.y = kg[kk * 4 + 1];
            acc = __builtin_amdgcn_wmma_f32_16x16x4_f32(false, aq[kk], false, b,
                                                        (short)0, acc, false, false);
        }
        float* ss = smem + LDS_S;        // C layout: VGPR r -> M = r + 8*half, N = l15
        #pragma unroll
        for (int r = 0; r < 8; ++r) {
            const int m = r + lh * 8;
            ss[m * SSTR + col0 + l15] = acc[r] * SCALE;
        }
    }
    __syncthreads();

    // ---- scores strip -> global: b128 non-temporal (write-once stream) ----
    {
        v4f* sg = (v4f*)(scores + (long long)bh * NTOK * NTOK + (long long)row0 * NTOK);
        #pragma unroll
        for (int i = 0; i < 16; ++i) {
            const int idx4 = tid + i * 256;          // 0..4095 float4s
            const int g = idx4 * 4;
            const int m = g >> 10, c = g & 1023;
            v4f t = *(const v4f*)(smem + LDS_S + m * SSTR + c);
            __builtin_nontemporal_store(t, sg + idx4);
        }
    }
    __syncthreads();

    // ---- softmax: wave w handles rows 2w, 2w+1 (wave32 reductions, float4 lanes) ----
    {
        float* ag = attn + (long long)bh * NTOK * NTOK + (long long)row0 * NTOK;
        #pragma unroll
        for (int rr = 0; rr < 2; ++rr) {
            const int m = w * 2 + rr;
            v4f* srow4 = (v4f*)(smem + LDS_S + m * SSTR);
            v4f vals[8];
            float mx = -1e30f;
            #pragma unroll
            for (int i = 0; i < 8; ++i) {
                vals[i] = srow4[lane + i * 32];
                mx = fmaxf(mx, fmaxf(fmaxf(vals[i].x, vals[i].y),
                                     fmaxf(vals[i].z, vals[i].w)));
            }
            mx = wave_max(mx);
            float sum = 0.f;
            #pragma unroll
            for (int i = 0; i < 8; ++i) {
                vals[i].x = __expf(vals[i].x - mx);
                vals[i].y = __expf(vals[i].y - mx);
                vals[i].z = __expf(vals[i].z - mx);
                vals[i].w = __expf(vals[i].w - mx);
                sum += (vals[i].x + vals[i].y) + (vals[i].z + vals[i].w);
            }
            const float inv = 1.0f / wave_sum(sum);
            v4f* ag4 = (v4f*)(ag + m * NTOK);
            #pragma unroll
            for (int i = 0; i < 8; ++i) {
                const v4f p = vals[i] * inv;
                srow4[lane + i * 32] = p;                       // reuse strip for attn @ V
                __builtin_nontemporal_store(p, ag4 + lane + i * 32);  // attn output (NT stream)
            }
        }
    }
    __syncthreads();

    // ---- context = attn @ V : wave w -> ctx col-tile (w&3), K half (w>>2) ----
    {
        const int nt = w & 3;
        const int kh = w >> 2;
        const float* vg = v + bhND + nt * 16 + l15;   // B[k][n] = V[k][16*nt+n]
        const float* arow = smem + LDS_S + l15 * SSTR + lh * 2;
        v8f acc = {};
        const int kbeg = kh * 512;
        #pragma unroll 4
        for (int k0 = kbeg; k0 < kbeg + 512; k0 += 4) {
            v2f a;
            a.x = arow[k0 + 0];
            a.y = arow[k0 + 1];
            v2f b;
            b.x = vg[(long long)(k0 + lh * 2 + 0) * DHEAD];
            b.y = vg[(long long)(k0 + lh * 2 + 1) * DHEAD];
            acc = __builtin_amdgcn_wmma_f32_16x16x4_f32(false, a, false, b,
                                                        (short)0, acc, false, false);
        }
        float* part = smem + LDS_P + w * 256;
        #pragma unroll
        for (int r = 0; r < 8; ++r) part[r * 32 + lane] = acc[r];
    }
    __syncthreads();

    // ---- reduce the two K-halves, write context (NT) ----
    if (w < 4) {
        const float* p0 = smem + LDS_P + w * 256;
        const float* p1 = smem + LDS_P + (w + 4) * 256;
        float* cg = ctx + bhND + (long long)row0 * DHEAD + w * 16;
        #pragma unroll
        for (int r = 0; r < 8; ++r) {
            const int m = r + lh * 8;
            __builtin_nontemporal_store(p0[r * 32 + lane] + p1[r * 32 + lane],
                                        &cg[(long long)m * DHEAD + l15]);
        }
    }
}

extern "C" void kernel_launch(void* const* d_in, const int* in_sizes, int n_in,
                              void* d_out, int out_size, void* d_ws, size_t ws_size,
                              hipStream_t stream) {
    const float* q = (const float*)d_in[0];
    const float* k = (const float*)d_in[1];
    const float* v = (const float*)d_in[2];
    float* out    = (float*)d_out;
    float* ctx    = out;                                              // [64,1024,64]
    float* scores = out + (long long)TBH * NTOK * DHEAD;              // [64,1024,1024]
    float* attn   = scores + (long long)TBH * NTOK * NTOK;            // [64,1024,1024]

    dim3 grid(NTOK / ROWS, TBH);   // 64 x 64 blocks
    dim3 block(256);               // 8 wave32s
    size_t smem = (size_t)LDS_FLOATS * sizeof(float);  // ~76.5 KB (CDNA5: 320KB/WGP)
    tempo_attn_kernel<<<grid, block, smem, stream>>>(q, k, v, ctx, scores, attn);
}